// ParabolicPool2D_TL_40106404610421
// MI455X (gfx1250) — compile-verified
//
#include <hip/hip_runtime.h>
#include <stdint.h>

// ParabolicPool2D: out[b,c,y,x] = max_{p,q in 3x3} f[b,c,2y+p,2x+q] + h[c,p,q]
// h[c] = -[[1,.5,1],[.5,0,.5],[1,.5,1]] * t[c]
//
// Memory-bound (AI ~0.85 flop/byte vs 23.3 TB/s HBM). Strategy:
//  - 1 workgroup per (b,c) plane; stage the whole 128x128 fp32 plane (64 KB)
//    into LDS with gfx1250 async global->LDS b128 loads (ASYNCcnt path),
//  - compute the stride-2 max-plus stencil from LDS (64 banks, stride-2
//    dword reads across 32 lanes are bank-conflict free),
//  - single coalesced b32 store per output element.

#define HH   128
#define WW   128
#define OH   63
#define OW   63
#define PLANE_IN  (HH * WW)   // 16384 floats = 64 KB
#define PLANE_OUT (OH * OW)   // 3969 floats
#define NTHREADS  256
#define CHUNKS    (PLANE_IN / 4 / NTHREADS)   // 16 float4 chunks per thread

__global__ __launch_bounds__(NTHREADS)
void parabolic_pool2d_kernel(const float* __restrict__ f,
                             const float* __restrict__ t,
                             float* __restrict__ out) {
    __shared__ float smem[PLANE_IN];

    const int plane = blockIdx.x;       // b*256 + c
    const int c     = plane & 255;
    const int tid   = threadIdx.x;

    // Block-uniform global base of this plane -> SGPR pair for GVS addressing.
    const float* gbase = f + (size_t)plane * PLANE_IN;

    // ---- Async load phase: global -> LDS, 128 bits per lane, coalesced ----
#pragma unroll
    for (int k = 0; k < CHUNKS; ++k) {
        const int idx4      = tid + k * NTHREADS;                   // float4 index
        const unsigned goff = (unsigned)(idx4 * 16);                // byte offset
        // Generic pointer to LDS: low 32 bits are the LDS byte address.
        const unsigned laddr = (unsigned)(uintptr_t)(&smem[idx4 * 4]);
        asm volatile("global_load_async_to_lds_b128 %0, %1, %2"
                     :
                     : "v"(laddr), "v"(goff), "s"(gbase)
                     : "memory");
    }

    // Per-channel stencil weights (uniform across the block -> scalar load).
    const float tc = t[c];
    const float w0 = -tc;          // corners (z = 1)
    const float w1 = -0.5f * tc;   // edges   (z = 0.5)
    // center weight is 0

    // __syncthreads() does not cover ASYNCcnt: wait for the async global->LDS loads,
    // then barrier so every wave sees the full plane in LDS.
#if __has_builtin(__builtin_amdgcn_s_wait_asynccnt)
    __builtin_amdgcn_s_wait_asynccnt(0);
#else
    asm volatile("s_wait_asynccnt 0x0" ::: "memory");
#endif
    __syncthreads();

    // ---- Compute phase: stride-2 3x3 max-plus from LDS ----
    float* optr = out + (size_t)plane * PLANE_OUT;
#pragma unroll 4
    for (int o = tid; o < PLANE_OUT; o += NTHREADS) {
        const int y = o / OW;
        const int x = o - y * OW;
        const float* r0 = &smem[(2 * y) * WW + 2 * x];
        const float* r1 = r0 + WW;
        const float* r2 = r1 + WW;

        float v;
        v = r0[0] + w0;
        v = fmaxf(v, r0[1] + w1);
        v = fmaxf(v, r0[2] + w0);
        v = fmaxf(v, r1[0] + w1);
        v = fmaxf(v, r1[1]);
        v = fmaxf(v, r1[2] + w1);
        v = fmaxf(v, r2[0] + w0);
        v = fmaxf(v, r2[1] + w1);
        v = fmaxf(v, r2[2] + w0);

        optr[o] = v;
    }
}

extern "C" void kernel_launch(void* const* d_in, const int* in_sizes, int n_in,
                              void* d_out, int out_size, void* d_ws, size_t ws_size,
                              hipStream_t stream) {
    const float* f = (const float*)d_in[0];   // (16, 256, 128, 128) fp32
    const float* t = (const float*)d_in[1];   // (256,) fp32
    float* out     = (float*)d_out;           // (16, 256, 63, 63) fp32

    const int nplanes = 16 * 256;             // one workgroup per (b, c) plane
    parabolic_pool2d_kernel<<<dim3(nplanes), dim3(NTHREADS), 0, stream>>>(f, t, out);
}